// GCN_25666724561211
// MI455X (gfx1250) — compile-verified
//
#include <hip/hip_runtime.h>

// ---------------------------------------------------------------------------
// GCN forward for MI455X (gfx1250, wave32).
//   h @ W      : fp32 WMMA (V_WMMA_F32_16X16X4_F32)
//                - block = 5 waves x 16 rows, shared (NT*16)-col N panel
//                - W k-panels staged to LDS via TDM (tensor_load_to_lds),
//                  double buffered, drained with s_wait_tensorcnt
//   aggregation: fused self-loop+bias init, then edge scatter w/ f32 atomics
//   ReLU fused into the next layer's WMMA A-fragment load
// ---------------------------------------------------------------------------

typedef __attribute__((ext_vector_type(2))) float v2f;
typedef __attribute__((ext_vector_type(8))) float v8f;
typedef __attribute__((ext_vector_type(4))) unsigned int u32x4;
typedef __attribute__((ext_vector_type(8))) int i32x8;
typedef __attribute__((ext_vector_type(4))) int i32x4;

#define N_NODES 50000
#define N_EDGES 800000
#define N_GRAPHS 64

// ------------------------- degree / norm kernels ---------------------------

__global__ void deg_init_kernel(float* __restrict__ deg, int n) {
    int i = blockIdx.x * blockDim.x + threadIdx.x;
    if (i < n) deg[i] = 1.0f;  // self-loop contribution
}

__global__ void deg_accum_kernel(const int* __restrict__ dst, float* __restrict__ deg, int e) {
    int i = blockIdx.x * blockDim.x + threadIdx.x;
    if (i < e) atomicAdd(&deg[dst[i]], 1.0f);
}

__global__ void deg_rsqrt_kernel(const float* __restrict__ deg, float* __restrict__ dis, int n) {
    int i = blockIdx.x * blockDim.x + threadIdx.x;
    if (i < n) {
        float d = deg[i];
        dis[i] = (d > 0.0f) ? rsqrtf(d) : 0.0f;
    }
}

// ---------------------- TDM descriptor issue (D#) --------------------------
// 2D tile of a row-major fp32 matrix: tile_rows x tile_cols at gptr, row
// stride = stride_elems. Groups per CDNA5 ISA.
__device__ __forceinline__ void tdm_load_w_tile(const float* gptr, unsigned lds_byte_off,
                                                int tile_cols, int tile_rows,
                                                int stride_elems, int tensor_rows) {
    unsigned long long ga = (unsigned long long)gptr;
    u32x4 g0;
    g0.x = 1u;                                                  // count=1, user mode
    g0.y = lds_byte_off;                                        // lds_addr
    g0.z = (unsigned)(ga & 0xFFFFFFFFu);                        // global_addr[31:0]
    g0.w = (unsigned)((ga >> 32) & 0x01FFFFFFu) | (2u << 30);   // addr[56:32] | type=2
    i32x8 g1;
    g1[0] = 0x20000;                                            // data_size=2 (4B), wg_mask=0
    g1[1] = (int)(((unsigned)stride_elems & 0xFFFFu) << 16);    // tensor_dim0[15:0] @ 63:48
    g1[2] = (int)((((unsigned)stride_elems >> 16) & 0xFFFFu) |  // tensor_dim0[31:16]
                  (((unsigned)tensor_rows & 0xFFFFu) << 16));   // tensor_dim1[15:0]
    g1[3] = (int)((((unsigned)tensor_rows >> 16) & 0xFFFFu) |   // tensor_dim1[31:16]
                  (((unsigned)tile_cols & 0xFFFFu) << 16));     // tile_dim0
    g1[4] = (int)((unsigned)tile_rows & 0xFFFFu);               // tile_dim1 (tile_dim2=0)
    g1[5] = stride_elems;                                       // tensor_dim0_stride[31:0]
    g1[6] = 0;                                                  // stride hi / dim1_stride lo
    g1[7] = 0;
    i32x4 z4 = {0, 0, 0, 0};
    i32x8 z8 = {0, 0, 0, 0, 0, 0, 0, 0};
    __builtin_amdgcn_tensor_load_to_lds(g0, g1, z4, z4, z8, 0);
}

// ------------------------------ WMMA GEMM ----------------------------------
// A fragment (16x4 fp32): lanes 0-15 hold M=lane, K = {0,1} in {v0,v1};
//                         lanes 16-31 hold M=lane-16, K = {2,3}.
// B fragment (4x16 fp32): mirrored K split across lane halves, N = lane&15.
// C/D (16x16 fp32): VGPR r -> row r (lanes 0-15) / row r+8 (lanes 16-31).

#define GEMM_WAVES 5   // 3125 row tiles = 625 blocks * 5 waves, exact
#define KB 64          // k-rows per LDS stage

template <int NT, bool RELU>
__global__ __launch_bounds__(GEMM_WAVES * 32) void gemm_wmma_tdm_kernel(
    const float* __restrict__ H, const float* __restrict__ W,
    float* __restrict__ M, int fin, int fout) {
    constexpr int TC = NT * 16;                 // panel cols
    __shared__ float smem[2 * KB * 64];         // 32 KB double buffer (max panel)

    const int lane  = threadIdx.x & 31;
    const int wave  = threadIdx.x >> 5;
    const int mlo   = lane & 15;
    const int khalf = (lane >> 4) << 1;         // 0 or 2
    const int tileM = blockIdx.x * GEMM_WAVES + wave;
    const int nbase = blockIdx.y * TC;
    const int row   = tileM * 16 + mlo;

    v8f acc[NT];
    const v8f zero = {};
#pragma unroll
    for (int t = 0; t < NT; ++t) acc[t] = zero;

    const float* __restrict__ hrow = H + (size_t)row * fin;
    const unsigned lds_base = (unsigned)(size_t)(&smem[0]);
    const int nstages = (fin + KB - 1) / KB;

    // prologue: stage 0
    if (wave == 0) {
        int r0 = fin < KB ? fin : KB;
        tdm_load_w_tile(W + nbase, lds_base, TC, r0, fout, fin);
        __builtin_amdgcn_s_wait_tensorcnt(0);
    }
    __syncthreads();

    for (int s = 0; s < nstages; ++s) {
        const int ks   = s * KB;
        const int rows = (fin - ks) < KB ? (fin - ks) : KB;

        // async prefetch of next stage into the other buffer
        if (wave == 0 && (s + 1) < nstages) {
            int rn = (fin - ks - KB) < KB ? (fin - ks - KB) : KB;
            tdm_load_w_tile(W + (size_t)(ks + KB) * fout + nbase,
                            lds_base + (unsigned)(((s + 1) & 1) * (KB * 64 * 4)),
                            TC, rn, fout, fin);
        }

        const float* __restrict__ bbuf = &smem[(s & 1) * (KB * 64)];
        for (int kk = 0; kk < rows; kk += 4) {
            float a0 = hrow[ks + kk + khalf];
            float a1 = hrow[ks + kk + khalf + 1];
            if (RELU) { a0 = fmaxf(a0, 0.0f); a1 = fmaxf(a1, 0.0f); }
            v2f a; a.x = a0; a.y = a1;
#pragma unroll
            for (int t = 0; t < NT; ++t) {
                v2f b;
                b.x = bbuf[(kk + khalf) * TC + t * 16 + mlo];
                b.y = bbuf[(kk + khalf + 1) * TC + t * 16 + mlo];
                acc[t] = __builtin_amdgcn_wmma_f32_16x16x4_f32(
                    false, a, false, b, (short)0, acc[t], false, false);
            }
        }

        __syncthreads();  // everyone done reading buf[s&1]
        if (wave == 0 && (s + 1) < nstages) __builtin_amdgcn_s_wait_tensorcnt(0);
        __syncthreads();  // next stage's LDS data visible to all waves
    }

    const int rbase = tileM * 16 + ((lane >> 4) ? 8 : 0);
#pragma unroll
    for (int t = 0; t < NT; ++t) {
        const int col = nbase + t * 16 + mlo;
#pragma unroll
        for (int r = 0; r < 8; ++r) {
            M[(size_t)(rbase + r) * fout + col] = acc[t][r];
        }
    }
}

// --------------------------- aggregation kernels ---------------------------

// agg[i][f] = m[i][f] * dis[i]^2 + bias[f]   (self-loop edge + bias)
__global__ void init_agg_kernel(const float* __restrict__ M, const float* __restrict__ dis,
                                const float* __restrict__ bias, float* __restrict__ A,
                                int n, int fout) {
    long long gid = (long long)blockIdx.x * blockDim.x + threadIdx.x;
    long long tot = (long long)n * fout;
    if (gid >= tot) return;
    int i = (int)(gid / fout);
    int f = (int)(gid % fout);
    float di = dis[i];
    A[gid] = M[gid] * di * di + bias[f];
}

// agg[dst][f] += m[src][f] * dis[src] * dis[dst]  over all edges; float4 per thread
__global__ void edge_scatter_kernel(const float* __restrict__ M, const int* __restrict__ src,
                                    const int* __restrict__ dst, const float* __restrict__ dis,
                                    float* __restrict__ A, int nedges, int fout) {
    const int chunks = fout >> 2;
    long long gid = (long long)blockIdx.x * blockDim.x + threadIdx.x;
    long long tot = (long long)nedges * chunks;
    if (gid >= tot) return;
    int e = (int)(gid / chunks);
    int c = (int)(gid % chunks) << 2;
    int s = src[e];
    int d = dst[e];
    float nrm = dis[s] * dis[d];
    const float4 mv = *(const float4*)(M + (size_t)s * fout + c);
    float* out = A + (size_t)d * fout + c;
    atomicAdd(out + 0, mv.x * nrm);
    atomicAdd(out + 1, mv.y * nrm);
    atomicAdd(out + 2, mv.z * nrm);
    atomicAdd(out + 3, mv.w * nrm);
}

// ------------------------------- pooling -----------------------------------

__global__ void pool_zero_kernel(float* __restrict__ sums, float* __restrict__ cnt) {
    int i = blockIdx.x * blockDim.x + threadIdx.x;
    if (i < N_GRAPHS * 16) sums[i] = 0.0f;
    if (i < N_GRAPHS) cnt[i] = 0.0f;
}

__global__ void pool_accum_kernel(const float* __restrict__ H, const int* __restrict__ batch,
                                  float* __restrict__ sums, float* __restrict__ cnt, int n) {
    long long gid = (long long)blockIdx.x * blockDim.x + threadIdx.x;
    if (gid >= (long long)n * 16) return;
    int i = (int)(gid >> 4);
    int f = (int)(gid & 15);
    int g = batch[i];
    atomicAdd(&sums[g * 16 + f], H[gid]);
    if (f == 0) atomicAdd(&cnt[g], 1.0f);
}

__global__ void pool_out_kernel(const float* __restrict__ sums, const float* __restrict__ cnt,
                                const float* __restrict__ Wc, const float* __restrict__ bc,
                                float* __restrict__ out) {
    int t = threadIdx.x;  // 0..191
    if (t >= N_GRAPHS * 3) return;
    int g = t / 3;
    int c = t % 3;
    float cc = fmaxf(cnt[g], 1.0f);
    float acc = bc[c];
#pragma unroll
    for (int f = 0; f < 16; ++f) acc += (sums[g * 16 + f] / cc) * Wc[f * 3 + c];
    out[t] = acc;
}

// ------------------------------- launcher ----------------------------------

extern "C" void kernel_launch(void* const* d_in, const int* in_sizes, int n_in,
                              void* d_out, int out_size, void* d_ws, size_t ws_size,
                              hipStream_t stream) {
    const float* x     = (const float*)d_in[0];
    const int*   edge  = (const int*)d_in[1];
    const int*   batch = (const int*)d_in[2];
    const float* Wl[6] = {(const float*)d_in[3],  (const float*)d_in[5],
                          (const float*)d_in[7],  (const float*)d_in[9],
                          (const float*)d_in[11], (const float*)d_in[13]};
    const float* bl[6] = {(const float*)d_in[4],  (const float*)d_in[6],
                          (const float*)d_in[8],  (const float*)d_in[10],
                          (const float*)d_in[12], (const float*)d_in[14]};
    const float* Wc = (const float*)d_in[15];
    const float* bc = (const float*)d_in[16];

    const int dims[7] = {1024, 512, 256, 128, 64, 32, 16};
    const int* src = edge;
    const int* dst = edge + N_EDGES;

    // workspace carve (max m/agg buffer = 50000*512*4 = 102,400,000 B)
    char* ws = (char*)d_ws;
    float* bufM = (float*)(ws);
    float* bufA = (float*)(ws + 102400000);
    float* bufB = (float*)(ws + 204800000);          // <= 51,200,000 B used
    float* deg  = (float*)(ws + 256000256);
    float* dis  = (float*)(ws + 256200448);
    float* sums = (float*)(ws + 256400640);
    float* cnt  = sums + N_GRAPHS * 16;

    // normalization: deg over (A + I) columns, dis = deg^-1/2
    deg_init_kernel<<<(N_NODES + 255) / 256, 256, 0, stream>>>(deg, N_NODES);
    deg_accum_kernel<<<(N_EDGES + 255) / 256, 256, 0, stream>>>(dst, deg, N_EDGES);
    deg_rsqrt_kernel<<<(N_NODES + 255) / 256, 256, 0, stream>>>(deg, dis, N_NODES);

    const int nBlocksM = (N_NODES / 16) / GEMM_WAVES;  // 3125 / 5 = 625 exact
    const float* hin = x;
    float* aggs[2] = {bufA, bufB};

    for (int i = 0; i < 6; ++i) {
        const int fin = dims[i], fout = dims[i + 1];
        float* m   = bufM;
        float* agg = aggs[i & 1];
        const bool relu = (i > 0);  // h1..h5 are ReLU'd inputs

        if (fout >= 64) {
            dim3 g(nBlocksM, fout / 64);
            if (relu) gemm_wmma_tdm_kernel<4, true><<<g, GEMM_WAVES * 32, 0, stream>>>(hin, Wl[i], m, fin, fout);
            else      gemm_wmma_tdm_kernel<4, false><<<g, GEMM_WAVES * 32, 0, stream>>>(hin, Wl[i], m, fin, fout);
        } else if (fout == 32) {
            dim3 g(nBlocksM, 1);
            gemm_wmma_tdm_kernel<2, true><<<g, GEMM_WAVES * 32, 0, stream>>>(hin, Wl[i], m, fin, fout);
        } else {  // fout == 16
            dim3 g(nBlocksM, 1);
            gemm_wmma_tdm_kernel<1, true><<<g, GEMM_WAVES * 32, 0, stream>>>(hin, Wl[i], m, fin, fout);
        }

        long long tot = (long long)N_NODES * fout;
        init_agg_kernel<<<(unsigned)((tot + 255) / 256), 256, 0, stream>>>(m, dis, bl[i], agg,
                                                                           N_NODES, fout);
        long long etot = (long long)N_EDGES * (fout >> 2);
        edge_scatter_kernel<<<(unsigned)((etot + 255) / 256), 256, 0, stream>>>(
            m, src, dst, dis, agg, N_EDGES, fout);

        hin = agg;
    }

    // global mean pool + classifier
    pool_zero_kernel<<<(N_GRAPHS * 16 + 255) / 256, 256, 0, stream>>>(sums, cnt);
    pool_accum_kernel<<<(N_NODES * 16 + 255) / 256, 256, 0, stream>>>(hin, batch, sums, cnt,
                                                                      N_NODES);
    pool_out_kernel<<<1, 192, 0, stream>>>(sums, cnt, Wc, bc, (float*)d_out);
}